// BertEmbeddings_83786222010462
// MI455X (gfx1250) — compile-verified
//
#include <hip/hip_runtime.h>
#include <cstdint>

#define HID 768
#define WPB 4          // waves (tokens) per block
#define NROWS 5        // gathered embedding rows per token

__device__ __forceinline__ float4 ld4(const float* p) { return *(const float4*)p; }

__global__ __launch_bounds__(32 * WPB, 1)
void bert_emb_fused(const int* __restrict__ word_ids,
                    const int* __restrict__ modal_ids,
                    const int* __restrict__ seg_ids,
                    const int* __restrict__ npi_ids,
                    const int* __restrict__ posi_ids,
                    const float* __restrict__ age_tau,
                    const float* __restrict__ delay_tau,
                    const float* __restrict__ word_tab,
                    const float* __restrict__ modal_tab,
                    const float* __restrict__ seg_tab,
                    const float* __restrict__ npi_tab,
                    const float* __restrict__ posi_tab,
                    const float* __restrict__ age_w,  const float* __restrict__ age_b,
                    const float* __restrict__ age_w0, const float* __restrict__ age_b0,
                    const float* __restrict__ del_w,  const float* __restrict__ del_b,
                    const float* __restrict__ del_w0, const float* __restrict__ del_b0,
                    const float* __restrict__ gamma,  const float* __restrict__ beta,
                    float* __restrict__ out, int n_tok)
{
    __shared__ __align__(16) float smem[WPB * NROWS * HID];   // 60 KB

    const int wave = threadIdx.x >> 5;
    const int lane = threadIdx.x & 31;
    const int t    = blockIdx.x * WPB + wave;
    if (t >= n_tok) return;                                   // whole wave uniform

    // ---- wave-uniform gather row bases -------------------------------------
    const float* rows[NROWS];
    rows[0] = word_tab  + (size_t)word_ids[t]  * HID;
    rows[1] = modal_tab + (size_t)modal_ids[t] * HID;
    rows[2] = seg_tab   + (size_t)seg_ids[t]   * HID;
    rows[3] = npi_tab   + (size_t)npi_ids[t]   * HID;
    rows[4] = posi_tab  + (size_t)posi_ids[t]  * HID;

    // LDS byte offset of this wave's staging slice (low 32 bits of the
    // flat-shared aperture address == workgroup-relative LDS address).
    const uint32_t sbase = (uint32_t)(uintptr_t)(&smem[wave * NROWS * HID]);

    // ---- CDNA5 async global->LDS DMA of the 5 gathered rows (3 KB each) ----
    // 6 chunks/row, b128/lane -> 512 B per instruction per wave; 30 in flight
    // on ASYNCcnt while we do the transcendental time2vec math below.
    #pragma unroll
    for (int r = 0; r < NROWS; ++r) {
        #pragma unroll
        for (int i = 0; i < 6; ++i) {
            const int e0 = i * 128 + lane * 4;
            const uint32_t dst = sbase + (uint32_t)((r * HID + e0) * sizeof(float));
            const float* src = rows[r] + e0;
            asm volatile("global_load_async_to_lds_b128 %0, %1, off"
                         :: "v"(dst), "v"(src) : "memory");
        }
    }

    // ---- time2vec (age + delay), overlapped with the async copies ----------
    const float ta  = age_tau[t];
    const float td  = delay_tau[t];
    const float aw0 = age_w0[0], ab0 = age_b0[0];
    const float dw0 = del_w0[0], db0 = del_b0[0];

    float4 x[6];
    #pragma unroll
    for (int i = 0; i < 6; ++i) {
        const int e0 = i * 128 + lane * 4;
        float v[4];
        if (e0 + 3 < HID - 1) {          // full vector: max index e0+3 <= 766
            const float4 aw = ld4(age_w + e0), ab = ld4(age_b + e0);
            const float4 dw = ld4(del_w + e0), db = ld4(del_b + e0);
            v[0] = __sinf(fmaf(ta, aw.x, ab.x)) + __sinf(fmaf(td, dw.x, db.x));
            v[1] = __sinf(fmaf(ta, aw.y, ab.y)) + __sinf(fmaf(td, dw.y, db.y));
            v[2] = __sinf(fmaf(ta, aw.z, ab.z)) + __sinf(fmaf(td, dw.z, db.z));
            v[3] = __sinf(fmaf(ta, aw.w, ab.w)) + __sinf(fmaf(td, dw.w, db.w));
        } else {                         // boundary tail: element 767 is linear
            #pragma unroll
            for (int c = 0; c < 4; ++c) {
                const int e = e0 + c;
                if (e < HID - 1)
                    v[c] = __sinf(fmaf(ta, age_w[e], age_b[e])) +
                           __sinf(fmaf(td, del_w[e], del_b[e]));
                else
                    v[c] = fmaf(ta, aw0, ab0) + fmaf(td, dw0, db0);
            }
        }
        x[i] = make_float4(v[0], v[1], v[2], v[3]);
    }

    // ---- wait for DMA, fold in the 5 gathered rows from LDS ----------------
    asm volatile("s_wait_asynccnt 0x0" ::: "memory");

    const float* sw = &smem[wave * NROWS * HID];
    float lsum = 0.f;
    #pragma unroll
    for (int i = 0; i < 6; ++i) {
        const int e0 = i * 128 + lane * 4;
        #pragma unroll
        for (int r = 0; r < NROWS; ++r) {
            const float4 e = *(const float4*)(sw + r * HID + e0);  // ds_load_b128
            x[i].x += e.x; x[i].y += e.y; x[i].z += e.z; x[i].w += e.w;
        }
        lsum += x[i].x + x[i].y + x[i].z + x[i].w;
    }

    // ---- LayerNorm over 768: wave32 cross-lane reductions ------------------
    #pragma unroll
    for (int off = 16; off > 0; off >>= 1) lsum += __shfl_xor(lsum, off, 32);
    const float mu = lsum * (1.0f / HID);

    float lvar = 0.f;
    #pragma unroll
    for (int i = 0; i < 6; ++i) {
        x[i].x -= mu; x[i].y -= mu; x[i].z -= mu; x[i].w -= mu;
        lvar = fmaf(x[i].x, x[i].x, lvar);
        lvar = fmaf(x[i].y, x[i].y, lvar);
        lvar = fmaf(x[i].z, x[i].z, lvar);
        lvar = fmaf(x[i].w, x[i].w, lvar);
    }
    #pragma unroll
    for (int off = 16; off > 0; off >>= 1) lvar += __shfl_xor(lvar, off, 32);
    const float rstd = rsqrtf(lvar * (1.0f / HID) + 1e-12f);

    // ---- scale/shift + 128-bit coalesced store -----------------------------
    float* op = out + (size_t)t * HID;
    #pragma unroll
    for (int i = 0; i < 6; ++i) {
        const int e0 = i * 128 + lane * 4;
        const float4 g = ld4(gamma + e0);
        const float4 b = ld4(beta  + e0);
        float4 y;
        y.x = fmaf(x[i].x * rstd, g.x, b.x);
        y.y = fmaf(x[i].y * rstd, g.y, b.y);
        y.z = fmaf(x[i].z * rstd, g.z, b.z);
        y.w = fmaf(x[i].w * rstd, g.w, b.w);
        *(float4*)(op + e0) = y;                               // global_store_b128
    }
}

extern "C" void kernel_launch(void* const* d_in, const int* in_sizes, int n_in,
                              void* d_out, int out_size, void* d_ws, size_t ws_size,
                              hipStream_t stream) {
    (void)n_in; (void)out_size; (void)d_ws; (void)ws_size;
    const int*   word_ids  = (const int*)  d_in[0];
    const int*   modal_ids = (const int*)  d_in[1];
    const int*   seg_ids   = (const int*)  d_in[2];
    const int*   npi_ids   = (const int*)  d_in[3];
    const int*   posi_ids  = (const int*)  d_in[4];
    const float* age_tau   = (const float*)d_in[5];
    const float* delay_tau = (const float*)d_in[6];
    const float* word_tab  = (const float*)d_in[7];
    const float* modal_tab = (const float*)d_in[8];
    const float* seg_tab   = (const float*)d_in[9];
    const float* npi_tab   = (const float*)d_in[10];
    const float* posi_tab  = (const float*)d_in[11];
    const float* age_w     = (const float*)d_in[12];
    const float* age_b     = (const float*)d_in[13];
    const float* age_w0    = (const float*)d_in[14];
    const float* age_b0    = (const float*)d_in[15];
    const float* del_w     = (const float*)d_in[16];
    const float* del_b     = (const float*)d_in[17];
    const float* del_w0    = (const float*)d_in[18];
    const float* del_b0    = (const float*)d_in[19];
    const float* ln_gamma  = (const float*)d_in[20];
    const float* ln_beta   = (const float*)d_in[21];

    const int n_tok  = in_sizes[0];                 // B*S = 16384
    const int blocks = (n_tok + WPB - 1) / WPB;     // 4096
    bert_emb_fused<<<blocks, 32 * WPB, 0, stream>>>(
        word_ids, modal_ids, seg_ids, npi_ids, posi_ids,
        age_tau, delay_tau,
        word_tab, modal_tab, seg_tab, npi_tab, posi_tab,
        age_w, age_b, age_w0, age_b0,
        del_w, del_b, del_w0, del_b0,
        ln_gamma, ln_beta,
        (float*)d_out, n_tok);
}